// Occupancy_68264210202776
// MI455X (gfx1250) — compile-verified
//
#include <hip/hip_runtime.h>
#include <hip/hip_bf16.h>

// ---------------------------------------------------------------------------
// Occupancy weights: per-ray gather + transmittance scan + reduce.
// Memory-bound (~516 MB HBM traffic -> ~22 us roofline @ 23.3 TB/s).
// Strategy:
//   * TDM (tensor_load_to_lds) async-DMAs each block's deltas/leaves tiles
//     into LDS (TENSORcnt + s_wait_tensorcnt) -- gfx1250 async-tensor path.
//   * wave32: one wave per ray; lane l owns samples [4l..4l+3] (B128 LDS
//     loads), cumprod via multiplicative Hillis-Steele wave scan (shuffles),
//     then wave reduction; lane 0 stores the ray result.
// ---------------------------------------------------------------------------

typedef unsigned int v4u __attribute__((ext_vector_type(4)));
typedef int          v8i __attribute__((ext_vector_type(8)));
typedef int          v4i __attribute__((ext_vector_type(4)));

#define RAYS_PER_BLOCK 8
#define NSAMP 128
#define TILE_ELEMS (RAYS_PER_BLOCK * NSAMP)   // 1024 elements = 4 KB

// Issue a 1-D TDM load: tile_elems contiguous 4-byte elements from gaddr
// into LDS byte offset lds_off. D# packed per CDNA5 ISA ch.8.
__device__ __forceinline__ void tdm_load_1d_b32(unsigned lds_off,
                                                const void* gaddr,
                                                unsigned tile_elems,
                                                unsigned tensor_elems) {
    unsigned long long ga = (unsigned long long)gaddr;

    // Group 0 (128b): count=1 | lds_addr | global_addr[56:0] | type=2
    v4u g0;
    g0.x = 1u;                                            // count=1, user mode
    g0.y = lds_off;                                       // lds_addr (bytes)
    g0.z = (unsigned)(ga & 0xFFFFFFFFu);                  // global_addr[31:0]
    g0.w = (unsigned)((ga >> 32) & 0x01FFFFFFu)           // global_addr[56:32]
         | (2u << 30);                                    // type=2 ("image")

    // Group 1 (256b): data_size=2 (4B); tensor_dim0; tile_dim0; stride0.
    v8i g1;
    g1[0] = (int)(2u << 16);                              // data_size = 4 bytes
    g1[1] = (int)((tensor_elems & 0xFFFFu) << 16);        // tensor_dim0[15:0] @bit48
    g1[2] = (int)(tensor_elems >> 16);                    // tensor_dim0[31:16]; dim1=0
    g1[3] = (int)((tile_elems & 0xFFFFu) << 16);          // tile_dim0 @bit112
    g1[4] = 0;                                            // tile_dim1=0, tile_dim2=0
    g1[5] = (int)tensor_elems;                            // tensor_dim0_stride[31:0]
    g1[6] = 0;                                            // stride hi bits
    g1[7] = 0;

    v4i g2 = {0, 0, 0, 0};                                // unused (<=2-D tensor)
    v4i g3 = {0, 0, 0, 0};
    v8i g4 = {0, 0, 0, 0, 0, 0, 0, 0};                    // extra group (clang-23 form)

    // amdgpu-toolchain (clang-23) 6-arg form.
    __builtin_amdgcn_tensor_load_to_lds(g0, g1, g2, g3, g4, /*cpol=*/0);
}

__global__ __launch_bounds__(256) void
occupancy_weights_kernel(const float* __restrict__ opacity,
                         const float* __restrict__ deltas,
                         const int*   __restrict__ leaves,
                         float* __restrict__ out,
                         int n_rays) {
    __shared__ __align__(16) float sD[TILE_ELEMS];
    __shared__ __align__(16) int   sL[TILE_ELEMS];

    const int tid  = threadIdx.x;
    const int wave = tid >> 5;
    const int lane = tid & 31;
    const long long r0 = (long long)blockIdx.x * RAYS_PER_BLOCK;

    // Wave 0 issues both tile DMAs (TDM ignores EXEC; wave-uniform branch
    // ensures a single issue per block), then drains TENSORcnt.
    if (wave == 0) {
        unsigned ldsD = (unsigned)(size_t)(void*)sD;   // low 32b of flat LDS addr = offset
        unsigned ldsL = (unsigned)(size_t)(void*)sL;
        unsigned total = (unsigned)n_rays * (unsigned)NSAMP;
        tdm_load_1d_b32(ldsD, deltas + r0 * NSAMP, TILE_ELEMS, total);
        tdm_load_1d_b32(ldsL, leaves + r0 * NSAMP, TILE_ELEMS, total);
        __builtin_amdgcn_s_wait_tensorcnt(0);
    }
    __syncthreads();

    const long long r = r0 + wave;
    if (r < (long long)n_rays) {
        // Lane l owns samples s = 4*lane .. 4*lane+3 of this wave's ray.
        const float4 d4 = ((const float4*)sD)[wave * 32 + lane];
        const int4   l4 = ((const int4*)sL)[wave * 32 + lane];

        // Gathers: opacity table is 2 MB -> resident in 192 MB L2.
        float a0 = 1.0f - __expf(-opacity[l4.x] * d4.x);
        float a1 = 1.0f - __expf(-opacity[l4.y] * d4.y);
        float a2 = 1.0f - __expf(-opacity[l4.z] * d4.z);
        float a3 = 1.0f - __expf(-opacity[l4.w] * d4.w);

        float t0 = fminf(1.0f, 1.0f - a0 + 1e-10f);
        float t1 = fminf(1.0f, 1.0f - a1 + 1e-10f);
        float t2 = fminf(1.0f, 1.0f - a2 + 1e-10f);
        float t3 = fminf(1.0f, 1.0f - a3 + 1e-10f);

        // Shifted cumprod: trans'[0] = 1 (first sample of the ray).
        if (lane == 0) t0 = 1.0f;

        // Local inclusive products within the lane's 4 samples.
        float p0 = t0;
        float p1 = p0 * t1;
        float p2 = p1 * t2;
        float p3 = p2 * t3;

        // Wave-wide multiplicative inclusive scan of lane totals (p3),
        // then shift to exclusive. 5 shuffle steps on wave32.
        float x = p3;
        #pragma unroll
        for (int off = 1; off < 32; off <<= 1) {
            float y = __shfl_up(x, off, 32);
            if (lane >= off) x *= y;
        }
        float excl = __shfl_up(x, 1, 32);
        if (lane == 0) excl = 1.0f;

        // weights[s] = alpha[s] * C[s], C[4l+j] = excl_l * p_j.
        float wsum = excl * (a0 * p0 + a1 * p1 + a2 * p2 + a3 * p3);

        // Wave reduction (sum) -> lane 0.
        #pragma unroll
        for (int off = 16; off > 0; off >>= 1)
            wsum += __shfl_down(wsum, off, 32);

        if (lane == 0) out[r] = wsum;
    }
}

extern "C" void kernel_launch(void* const* d_in, const int* in_sizes, int n_in,
                              void* d_out, int out_size, void* d_ws, size_t ws_size,
                              hipStream_t stream) {
    const float* opacity = (const float*)d_in[0];
    const float* deltas  = (const float*)d_in[1];
    const int*   leaves  = (const int*)d_in[2];   // jnp int64 downcast to i32 (x64 off)
    float* out = (float*)d_out;

    const int n_rays = in_sizes[1] / NSAMP;       // deltas is [n_rays, 128]
    const int blocks = (n_rays + RAYS_PER_BLOCK - 1) / RAYS_PER_BLOCK;

    occupancy_weights_kernel<<<blocks, 256, 0, stream>>>(opacity, deltas, leaves,
                                                         out, n_rays);
}